// CMultiheadAttention_28862180229412
// MI455X (gfx1250) — compile-verified
//
#include <hip/hip_runtime.h>
#include <cstdint>

// ---------------------------------------------------------------------------
// Complex multihead attention for MI455X (gfx1250), fp32 via V_WMMA_F32_16X16X4_F32
// T=1024, B=4, E=1024, H=16, HD=64
// Async global->LDS staging (ASYNCcnt) with double-buffered software pipeline.
// ---------------------------------------------------------------------------

typedef __attribute__((ext_vector_type(2))) float v2f;
typedef __attribute__((ext_vector_type(8))) float v8f;

#define T_DIM 1024
#define B_DIM 4
#define E_DIM 1024
#define H_DIM 16
#define HD_DIM 64
#define M_DIM 4096          // T*B
#define N3E 3072            // 3*E
#define QK_SCALE 0.125f     // HD^-0.5

__device__ __forceinline__ v8f zero8() {
  v8f z = {0.f, 0.f, 0.f, 0.f, 0.f, 0.f, 0.f, 0.f};
  return z;
}

// D = A(16x4) x B(4x16) + C, fp32
__device__ __forceinline__ v8f wmma4(v2f a, v2f b, v8f c) {
  return __builtin_amdgcn_wmma_f32_16x16x4_f32(false, a, false, b, (short)0, c,
                                               false, false);
}

// A-fragment (and N-major B-fragment): 32-bit 16x4 layout
__device__ __forceinline__ v2f fragA(const float* s, int stride, int m, int k, int hi) {
  v2f r;
  r.x = s[m * stride + k + 2 * hi];
  r.y = s[m * stride + k + 1 + 2 * hi];
  return r;
}

// B-fragment from K-major LDS tile (rows = K, cols = N)
__device__ __forceinline__ v2f fragBk(const float* s, int stride, int k, int n, int hi) {
  v2f r;
  r.x = s[(k + 2 * hi) * stride + n];
  r.y = s[(k + 1 + 2 * hi) * stride + n];
  return r;
}

__device__ __forceinline__ v2f negf2(v2f a) {
  v2f r; r.x = -a.x; r.y = -a.y; return r;
}

__device__ __forceinline__ void st4(float* dst, float4 v) {
  dst[0] = v.x; dst[1] = v.y; dst[2] = v.z; dst[3] = v.w;
}

// Async global -> LDS copy, 16 bytes per lane, tracked by ASYNCcnt.
// LDS byte address = low 32 bits of the generic pointer (aperture: addr[31:0]).
__device__ __forceinline__ void async_ld_b128(const float* lds_dst, const float* g) {
  uint32_t lds_off = (uint32_t)(uintptr_t)lds_dst;
  asm volatile("global_load_async_to_lds_b128 %0, %1, off"
               :: "v"(lds_off), "v"(g)
               : "memory");
}

__device__ __forceinline__ void wait_async0() {
#if __has_builtin(__builtin_amdgcn_s_wait_asynccnt)
  __builtin_amdgcn_s_wait_asynccnt(0);
#else
  asm volatile("s_wait_asynccnt 0x0" ::: "memory");
#endif
}

// ---------------------------------------------------------------------------
// Complex NT-GEMM:  C[m,n] = sum_k A[m,k]*B[n,k] (complex) + bias
// 64x64 block tile, 8 waves; double-buffered async LDS staging.
// ---------------------------------------------------------------------------
__global__ __launch_bounds__(256) void cgemm_nt_kernel(
    const float* __restrict__ Are, const float* __restrict__ Aim,
    const float* __restrict__ Bre, const float* __restrict__ Bim,
    const float* __restrict__ biasre, const float* __restrict__ biasim,
    float* __restrict__ Cre, float* __restrict__ Cim,
    int N, int K) {
  __shared__ float sAre[2][64][17], sAim[2][64][17];
  __shared__ float sBre[2][64][17], sBim[2][64][17];

  const int tid  = threadIdx.x;
  const int lane = tid & 31;
  const int w    = tid >> 5;
  const int wm   = w >> 2;        // 0..1
  const int wn   = w & 3;         // 0..3
  const int lm   = lane & 15;
  const int hi   = lane >> 4;
  const int m0g  = blockIdx.y * 64;
  const int n0g  = blockIdx.x * 64;

  v8f accre[2], accim[2];
  accre[0] = zero8(); accre[1] = zero8();
  accim[0] = zero8(); accim[1] = zero8();

  const int lrow = tid >> 2;        // 0..63
  const int lcol = (tid & 3) * 4;   // 0,4,8,12

  auto issue = [&](int buf, int kb) {
    const size_t aoff = (size_t)(m0g + lrow) * K + kb + lcol;
    const size_t boff = (size_t)(n0g + lrow) * K + kb + lcol;
    async_ld_b128(&sAre[buf][lrow][lcol], Are + aoff);
    async_ld_b128(&sAim[buf][lrow][lcol], Aim + aoff);
    async_ld_b128(&sBre[buf][lrow][lcol], Bre + boff);
    async_ld_b128(&sBim[buf][lrow][lcol], Bim + boff);
  };

  issue(0, 0);

  for (int kb = 0; kb < K; kb += 16) {
    const int buf = (kb >> 4) & 1;
    wait_async0();          // this wave's async writes for `buf` are in LDS
    __syncthreads();        // everyone's writes are in LDS; prior reads done
    if (kb + 16 < K) issue(buf ^ 1, kb + 16);  // overlap next tile with compute

    const float* pAre = &sAre[buf][0][0];
    const float* pAim = &sAim[buf][0][0];
    const float* pBre = &sBre[buf][0][0];
    const float* pBim = &sBim[buf][0][0];

#pragma unroll
    for (int ks = 0; ks < 16; ks += 4) {
      v2f br = fragA(pBre, 17, wn * 16 + lm, ks, hi);
      v2f bi = fragA(pBim, 17, wn * 16 + lm, ks, hi);
#pragma unroll
      for (int s = 0; s < 2; ++s) {
        const int m = wm * 32 + s * 16 + lm;
        v2f ar = fragA(pAre, 17, m, ks, hi);
        v2f ai = fragA(pAim, 17, m, ks, hi);
        accre[s] = wmma4(ar, br, accre[s]);
        accre[s] = wmma4(negf2(ai), bi, accre[s]);   // -Aim*Bim
        accim[s] = wmma4(ar, bi, accim[s]);
        accim[s] = wmma4(ai, br, accim[s]);
      }
    }
  }

  const int ng = n0g + wn * 16 + lm;
  const float bre = biasre[ng];
  const float bim = biasim[ng];
#pragma unroll
  for (int s = 0; s < 2; ++s) {
#pragma unroll
    for (int r = 0; r < 8; ++r) {
      const int m = m0g + wm * 32 + s * 16 + r + 8 * hi;
      Cre[(size_t)m * N + ng] = accre[s][r] + bre;
      Cim[(size_t)m * N + ng] = accim[s][r] + bim;
    }
  }
}

// ---------------------------------------------------------------------------
// awr[bh, tq, tk] = QK_SCALE * ( qr.(kr+ki) + qi.(kr-ki) ) ; K = HD = 64
// (sum/diff transform needs VALU, so staging stays manual here)
// ---------------------------------------------------------------------------
__global__ __launch_bounds__(256) void qk_kernel(
    const float* __restrict__ Pre, const float* __restrict__ Pim,
    float* __restrict__ awr) {
  __shared__ float sQr[64][33], sQi[64][33], sKs[64][33], sKd[64][33];

  const int tid  = threadIdx.x;
  const int lane = tid & 31;
  const int w    = tid >> 5;
  const int wm   = w >> 2;
  const int wn   = w & 3;
  const int lm   = lane & 15;
  const int hi   = lane >> 4;
  const int bh   = blockIdx.z;
  const int b    = bh >> 4;
  const int h    = bh & 15;
  const int tq0  = blockIdx.y * 64;
  const int tk0  = blockIdx.x * 64;

  v8f acc[2];
  acc[0] = zero8(); acc[1] = zero8();

  for (int kb = 0; kb < HD_DIM; kb += 32) {
#pragma unroll
    for (int i = 0; i < 2; ++i) {
      const int p   = tid + i * 256;   // 0..511 float4 slots
      const int row = p >> 3;          // 0..63
      const int col = (p & 7) * 4;     // 0..28
      const size_t qoff =
          ((size_t)(tq0 + row) * B_DIM + b) * N3E + h * HD_DIM + kb + col;
      st4(&sQr[row][col], *(const float4*)(Pre + qoff));
      st4(&sQi[row][col], *(const float4*)(Pim + qoff));
      const size_t koff =
          ((size_t)(tk0 + row) * B_DIM + b) * N3E + E_DIM + h * HD_DIM + kb + col;
      const float4 kr = *(const float4*)(Pre + koff);
      const float4 ki = *(const float4*)(Pim + koff);
      st4(&sKs[row][col], make_float4(kr.x + ki.x, kr.y + ki.y, kr.z + ki.z, kr.w + ki.w));
      st4(&sKd[row][col], make_float4(kr.x - ki.x, kr.y - ki.y, kr.z - ki.z, kr.w - ki.w));
    }
    __syncthreads();

#pragma unroll
    for (int ks = 0; ks < 32; ks += 4) {
      v2f bs = fragA(&sKs[0][0], 33, wn * 16 + lm, ks, hi);
      v2f bd = fragA(&sKd[0][0], 33, wn * 16 + lm, ks, hi);
#pragma unroll
      for (int s = 0; s < 2; ++s) {
        const int m = wm * 32 + s * 16 + lm;
        v2f qr = fragA(&sQr[0][0], 33, m, ks, hi);
        v2f qi = fragA(&sQi[0][0], 33, m, ks, hi);
        acc[s] = wmma4(qr, bs, acc[s]);
        acc[s] = wmma4(qi, bd, acc[s]);
      }
    }
    __syncthreads();
  }

  const int tk = tk0 + wn * 16 + lm;
#pragma unroll
  for (int s = 0; s < 2; ++s) {
#pragma unroll
    for (int r = 0; r < 8; ++r) {
      const int tq = tq0 + wm * 32 + s * 16 + r + 8 * hi;
      awr[((size_t)bh << 20) + (size_t)tq * 1024 + tk] = QK_SCALE * acc[s][r];
    }
  }
}

// ---------------------------------------------------------------------------
// global min/max over awr (two-stage deterministic reduction)
// ---------------------------------------------------------------------------
__global__ __launch_bounds__(256) void minmax_part_kernel(
    const float* __restrict__ awr, float* __restrict__ pmin,
    float* __restrict__ pmax) {
  const float4* a4 = (const float4*)awr;
  const size_t n4 = (size_t)64 * 1024 * 1024 / 4;  // 16,777,216 float4
  float mn = 3.402823466e38f, mx = -3.402823466e38f;
  for (size_t i = (size_t)blockIdx.x * blockDim.x + threadIdx.x; i < n4;
       i += (size_t)gridDim.x * blockDim.x) {
    const float4 v = a4[i];
    mn = fminf(mn, fminf(fminf(v.x, v.y), fminf(v.z, v.w)));
    mx = fmaxf(mx, fmaxf(fmaxf(v.x, v.y), fmaxf(v.z, v.w)));
  }
  __shared__ float smn[256], smx[256];
  const int t = threadIdx.x;
  smn[t] = mn; smx[t] = mx;
  __syncthreads();
  for (int s = 128; s > 0; s >>= 1) {
    if (t < s) {
      smn[t] = fminf(smn[t], smn[t + s]);
      smx[t] = fmaxf(smx[t], smx[t + s]);
    }
    __syncthreads();
  }
  if (t == 0) { pmin[blockIdx.x] = smn[0]; pmax[blockIdx.x] = smx[0]; }
}

__global__ __launch_bounds__(256) void minmax_final_kernel(
    const float* __restrict__ pmin, const float* __restrict__ pmax,
    float* __restrict__ stats) {
  __shared__ float smn[256], smx[256];
  const int t = threadIdx.x;
  float mn = 3.402823466e38f, mx = -3.402823466e38f;
  for (int i = t; i < 1024; i += 256) {
    mn = fminf(mn, pmin[i]);
    mx = fmaxf(mx, pmax[i]);
  }
  smn[t] = mn; smx[t] = mx;
  __syncthreads();
  for (int s = 128; s > 0; s >>= 1) {
    if (t < s) {
      smn[t] = fminf(smn[t], smn[t + s]);
      smx[t] = fmaxf(smx[t], smx[t + s]);
    }
    __syncthreads();
  }
  if (t == 0) {
    stats[0] = smn[0];
    stats[1] = 1.0f / (smx[0] - smn[0]);
  }
}

// ---------------------------------------------------------------------------
// attn[t,b,h*64+d] = sum_tk norm(awr[bh,tq,tk]) * v[tk,d]  (re & im)
// A tile normalized via VALU (manual staging); V tiles via async LDS loads.
// ---------------------------------------------------------------------------
__global__ __launch_bounds__(256) void av_kernel(
    const float* __restrict__ Pre, const float* __restrict__ Pim,
    const float* __restrict__ awr, const float* __restrict__ stats,
    float* __restrict__ Atre, float* __restrict__ Atim) {
  __shared__ float sA[64][65], sVr[64][65], sVi[64][65];

  const int tid  = threadIdx.x;
  const int lane = tid & 31;
  const int w    = tid >> 5;
  const int wm   = w >> 2;
  const int wn   = w & 3;
  const int lm   = lane & 15;
  const int hi   = lane >> 4;
  const int bh   = blockIdx.y;
  const int b    = bh >> 4;
  const int h    = bh & 15;
  const int tq0  = blockIdx.x * 64;
  const float mn  = stats[0];
  const float inv = stats[1];

  v8f accre[2], accim[2];
  accre[0] = zero8(); accre[1] = zero8();
  accim[0] = zero8(); accim[1] = zero8();

  for (int kb = 0; kb < 1024; kb += 64) {
#pragma unroll
    for (int i = 0; i < 4; ++i) {
      const int p   = tid + i * 256;  // 0..1023 float4 slots
      const int row = p >> 4;         // 0..63
      const int col = (p & 15) * 4;   // 0..60
      // async V tiles (raw copy -> async engine)
      const size_t voff =
          ((size_t)(kb + row) * B_DIM + b) * N3E + 2 * E_DIM + h * HD_DIM + col;
      async_ld_b128(&sVr[row][col], Pre + voff);
      async_ld_b128(&sVi[row][col], Pim + voff);
      // normalized awr tile (needs VALU -> manual staging)
      const float4 a4 = *(const float4*)(awr + ((size_t)bh << 20) +
                                         (size_t)(tq0 + row) * 1024 + kb + col);
      st4(&sA[row][col], make_float4((a4.x - mn) * inv, (a4.y - mn) * inv,
                                     (a4.z - mn) * inv, (a4.w - mn) * inv));
    }
    wait_async0();
    __syncthreads();

#pragma unroll
    for (int ks = 0; ks < 64; ks += 4) {
      v2f bvr = fragBk(&sVr[0][0], 65, ks, wn * 16 + lm, hi);
      v2f bvi = fragBk(&sVi[0][0], 65, ks, wn * 16 + lm, hi);
#pragma unroll
      for (int s = 0; s < 2; ++s) {
        v2f aa = fragA(&sA[0][0], 65, wm * 32 + s * 16 + lm, ks, hi);
        accre[s] = wmma4(aa, bvr, accre[s]);
        accim[s] = wmma4(aa, bvi, accim[s]);
      }
    }
    __syncthreads();
  }

  const int d = wn * 16 + lm;
#pragma unroll
  for (int s = 0; s < 2; ++s) {
#pragma unroll
    for (int r = 0; r < 8; ++r) {
      const int tq = tq0 + wm * 32 + s * 16 + r + 8 * hi;
      const size_t off = ((size_t)tq * B_DIM + b) * E_DIM + h * HD_DIM + d;
      Atre[off] = accre[s][r];
      Atim[off] = accim[s][r];
    }
  }
}

// ---------------------------------------------------------------------------
// aw_avg[b,tq,tk] = ( mean_h awr[b*16+h,tq,tk] - mn ) * inv
// ---------------------------------------------------------------------------
__global__ __launch_bounds__(256) void awavg_kernel(
    const float* __restrict__ awr, const float* __restrict__ stats,
    float* __restrict__ out) {
  const float mn  = stats[0];
  const float inv = stats[1];
  const size_t n = (size_t)B_DIM * 1024 * 1024;  // 4,194,304
  for (size_t idx = (size_t)blockIdx.x * blockDim.x + threadIdx.x; idx < n;
       idx += (size_t)gridDim.x * blockDim.x) {
    const size_t b   = idx >> 20;
    const size_t rem = idx & 0xFFFFF;
    float s = 0.f;
#pragma unroll
    for (int h = 0; h < H_DIM; ++h)
      s += awr[(((size_t)b * H_DIM + h) << 20) + rem];
    out[idx] = (s * (1.0f / H_DIM) - mn) * inv;
  }
}

// ---------------------------------------------------------------------------
// Host-side launch
// ---------------------------------------------------------------------------
extern "C" void kernel_launch(void* const* d_in, const int* in_sizes, int n_in,
                              void* d_out, int out_size, void* d_ws,
                              size_t ws_size, hipStream_t stream) {
  (void)in_sizes; (void)n_in; (void)out_size; (void)ws_size;
  const float* x_re  = (const float*)d_in[0];
  const float* x_im  = (const float*)d_in[1];
  const float* w_re  = (const float*)d_in[2];
  const float* w_im  = (const float*)d_in[3];
  const float* b_re  = (const float*)d_in[4];
  const float* b_im  = (const float*)d_in[5];
  const float* ow_re = (const float*)d_in[6];
  const float* ow_im = (const float*)d_in[7];
  const float* ob_re = (const float*)d_in[8];
  const float* ob_im = (const float*)d_in[9];

  // workspace layout (floats); total ~403 MB
  float* ws      = (float*)d_ws;
  float* proj_re = ws;                                   // 4096*3072
  float* proj_im = proj_re + (size_t)M_DIM * N3E;        // 4096*3072
  float* awr     = proj_im + (size_t)M_DIM * N3E;        // 64*1024*1024
  float* attn_re = awr + (size_t)64 * 1024 * 1024;       // 4096*1024
  float* attn_im = attn_re + (size_t)M_DIM * E_DIM;      // 4096*1024
  float* pmin    = attn_im + (size_t)M_DIM * E_DIM;      // 1024
  float* pmax    = pmin + 1024;                          // 1024
  float* stats   = pmax + 1024;                          // 2

  float* out_re = (float*)d_out;                         // (T,B,E)
  float* out_im = out_re + (size_t)M_DIM * E_DIM;        // (T,B,E)
  float* aw_avg = out_im + (size_t)M_DIM * E_DIM;        // (B,T,T)

  // 1) in_proj: proj = x @ W^T + b  (complex), M=4096 N=3072 K=1024
  cgemm_nt_kernel<<<dim3(N3E / 64, M_DIM / 64), 256, 0, stream>>>(
      x_re, x_im, w_re, w_im, b_re, b_im, proj_re, proj_im, N3E, E_DIM);

  // 2) awr = scale * (qr.(kr+ki) + qi.(kr-ki)) per head-batch
  qk_kernel<<<dim3(16, 16, 64), 256, 0, stream>>>(proj_re, proj_im, awr);

  // 3) global min/max -> stats {mn, 1/(mx-mn)}
  minmax_part_kernel<<<1024, 256, 0, stream>>>(awr, pmin, pmax);
  minmax_final_kernel<<<1, 256, 0, stream>>>(pmin, pmax, stats);

  // 4) attn = norm(awr) @ v  (re & im), written in (T,B,E) layout
  av_kernel<<<dim3(16, 64), 256, 0, stream>>>(proj_re, proj_im, awr, stats,
                                              attn_re, attn_im);

  // 5) aw_avg output
  awavg_kernel<<<4096, 256, 0, stream>>>(awr, stats, aw_avg);

  // 6) out_proj: out = attn @ OW^T + ob (complex), M=4096 N=1024 K=1024
  cgemm_nt_kernel<<<dim3(E_DIM / 64, M_DIM / 64), 256, 0, stream>>>(
      attn_re, attn_im, ow_re, ow_im, ob_re, ob_im, out_re, out_im, E_DIM,
      E_DIM);
}